// GCN_17970143166990
// MI455X (gfx1250) — compile-verified
//
#include <hip/hip_runtime.h>
#include <hip/hip_bf16.h>

#define DF 128          // D == H == 128
#define SA_STRIDE 132   // padded LDS row stride (banks: stride%64=4 -> conflict-free)

typedef float v2f __attribute__((ext_vector_type(2)));
typedef float v8f __attribute__((ext_vector_type(8)));

// ---------------------------------------------------------------------------
// O[n x 128] = act(A[n x 128]) @ W[128 x 128], fp32 WMMA 16x16x4.
// One wave per 16x16 output tile; block = 8 waves = one 16-row strip.
// ---------------------------------------------------------------------------
__global__ __launch_bounds__(256)
void gemm_wmma_f32(const float* __restrict__ A, const float* __restrict__ W,
                   float* __restrict__ O, int nrows, int relu_in) {
    __shared__ float sA[16 * SA_STRIDE];
    const int tid  = threadIdx.x;
    const int row0 = blockIdx.x * 16;

    // cooperative load of 16x128 A tile with optional ReLU
    for (int i = tid; i < 16 * DF; i += 256) {
        const int r = i >> 7;           // 0..15
        const int cix = i & (DF - 1);   // 0..127
        float v = 0.f;
        if (row0 + r < nrows) {
            v = A[(size_t)(row0 + r) * DF + cix];
            if (relu_in) v = fmaxf(v, 0.f);
        }
        sA[r * SA_STRIDE + cix] = v;
    }
    __syncthreads();

    const int wave = tid >> 5;      // 0..7 -> column tile
    const int lane = tid & 31;
    const int col0 = wave * 16;
    const int mn   = lane & 15;     // A-row / B-col for this lane
    const int half = lane >> 4;     // K-pair select: {0,1} vs {2,3}

    v8f c = {0.f, 0.f, 0.f, 0.f, 0.f, 0.f, 0.f, 0.f};
    for (int k = 0; k < DF; k += 4) {
        const int kb = k + 2 * half;
        v2f a, b;
        a.x = sA[mn * SA_STRIDE + kb];
        a.y = sA[mn * SA_STRIDE + kb + 1];
        b.x = W[(size_t)kb * DF + col0 + mn];
        b.y = W[(size_t)(kb + 1) * DF + col0 + mn];
        c = __builtin_amdgcn_wmma_f32_16x16x4_f32(false, a, false, b,
                                                  (short)0, c, false, false);
    }

    // C/D layout: VGPR i -> M = i + 8*half, N = mn
    for (int i = 0; i < 8; ++i) {
        const int mm = i + 8 * half;
        if (row0 + mm < nrows)
            O[(size_t)(row0 + mm) * DF + col0 + mn] = c[i];
    }
}

// ---------------------------------------------------------------------------
// degree / normalization
// ---------------------------------------------------------------------------
__global__ void deg_kernel(const int* __restrict__ dst, float* __restrict__ deg, int E) {
    int e = blockIdx.x * blockDim.x + threadIdx.x;
    if (e < E) atomicAdd(deg + dst[e], 1.0f);
}

__global__ void dinv_kernel(float* __restrict__ deg, int n) {
    int i = blockIdx.x * blockDim.x + threadIdx.x;
    if (i < n) deg[i] = rsqrtf(deg[i] + 1.0f);   // +1 self-loop
}

// agg = h * dinv^2  (self-loop term seeds the accumulator)
__global__ void init_agg(const float* __restrict__ h, const float* __restrict__ dinv,
                         float* __restrict__ agg, int n) {
    int t = blockIdx.x * blockDim.x + threadIdx.x;   // n*32 float4 slots
    if (t >= n * 32) return;
    const int node = t >> 5, q = t & 31;
    const float di = dinv[node];
    const float w = di * di;
    float4 v = ((const float4*)h)[(size_t)node * 32 + q];
    v.x *= w; v.y *= w; v.z *= w; v.w *= w;
    ((float4*)agg)[(size_t)node * 32 + q] = v;
}

// one wave per edge: agg[dst] += h[src] * dinv[src]*dinv[dst]
__global__ void edge_agg(const float* __restrict__ h, const int* __restrict__ src,
                         const int* __restrict__ dst, const float* __restrict__ dinv,
                         float* __restrict__ agg, int E) {
    int t = blockIdx.x * blockDim.x + threadIdx.x;
    int e = t >> 5;
    if (e >= E) return;
    const int lane = t & 31;
    const int s = src[e], d = dst[e];
    const float w = dinv[s] * dinv[d];
    const float4 v = *(const float4*)(h + (size_t)s * DF + lane * 4);
    float* out = agg + (size_t)d * DF + lane * 4;
    atomicAdd(out + 0, v.x * w);
    atomicAdd(out + 1, v.y * w);
    atomicAdd(out + 2, v.z * w);
    atomicAdd(out + 3, v.w * w);
}

// ---------------------------------------------------------------------------
// global mean pool
// ---------------------------------------------------------------------------
__global__ void pool_accum(const float* __restrict__ h, const int* __restrict__ batch,
                           float* __restrict__ pooled, float* __restrict__ cnt, int n) {
    int t = blockIdx.x * blockDim.x + threadIdx.x;
    if (t >= n * 32) return;
    const int node = t >> 5, q = t & 31;
    const int g = batch[node];
    const float4 v = ((const float4*)h)[(size_t)node * 32 + q];
    float* p = pooled + (size_t)g * DF + q * 4;
    atomicAdd(p + 0, v.x);
    atomicAdd(p + 1, v.y);
    atomicAdd(p + 2, v.z);
    atomicAdd(p + 3, v.w);
    if (q == 0) atomicAdd(cnt + g, 1.0f);
}

// ---------------------------------------------------------------------------
// MLP head (tiny)
// ---------------------------------------------------------------------------
__global__ void mlp1_kernel(const float* __restrict__ pooled, const float* __restrict__ cnt,
                            const float* __restrict__ M0w, const float* __restrict__ M0b,
                            float* __restrict__ out, int G) {
    int t = blockIdx.x * blockDim.x + threadIdx.x;
    if (t >= G * DF) return;
    const int g = t / DF, j = t % DF;
    const float inv = 1.0f / fmaxf(cnt[g], 1.0f);
    float acc = M0b[j];
    for (int k = 0; k < DF; ++k)
        acc += pooled[(size_t)g * DF + k] * inv * M0w[(size_t)k * DF + j];
    out[t] = fmaxf(acc, 0.f);
}

__global__ void mlp2_kernel(const float* __restrict__ hidden, const float* __restrict__ M1w,
                            const float* __restrict__ M1b, float* __restrict__ out,
                            int G, int OUT) {
    int t = blockIdx.x * blockDim.x + threadIdx.x;
    if (t >= G * OUT) return;
    const int g = t / OUT, o = t % OUT;
    float acc = M1b[o];
    for (int j = 0; j < DF; ++j)
        acc += hidden[(size_t)g * DF + j] * M1w[(size_t)j * OUT + o];
    out[t] = acc;
}

// ---------------------------------------------------------------------------
extern "C" void kernel_launch(void* const* d_in, const int* in_sizes, int n_in,
                              void* d_out, int out_size, void* d_ws, size_t ws_size,
                              hipStream_t stream) {
    const float* x   = (const float*)d_in[0];
    const float* W0  = (const float*)d_in[1];
    const float* W1  = (const float*)d_in[2];
    const float* W2  = (const float*)d_in[3];
    const float* M0w = (const float*)d_in[4];
    const float* M0b = (const float*)d_in[5];
    const float* M1w = (const float*)d_in[6];
    const float* M1b = (const float*)d_in[7];
    const int*   ei  = (const int*)d_in[8];
    const int*   bat = (const int*)d_in[9];

    const int N   = in_sizes[0] / DF;
    const int E   = in_sizes[8] / 2;
    const int OUT = in_sizes[7];
    const int G   = out_size / OUT;
    const int* src = ei;
    const int* dst = ei + E;

    // workspace layout (floats)
    float* ws = (float*)d_ws;
    size_t o = 0;
    float* dinv   = ws + o;  o += ((size_t)N + 63) & ~(size_t)63;
    float* buf0   = ws + o;  o += (size_t)N * DF;
    float* buf1   = ws + o;  o += (size_t)N * DF;
    float* pooled = ws + o;  o += (size_t)G * DF;
    float* cnt    = ws + o;  o += ((size_t)G + 63) & ~(size_t)63;
    float* mlp1   = ws + o;  o += (size_t)G * DF;
    (void)ws_size; (void)n_in;

    // normalization (same for all 3 layers)
    hipMemsetAsync(dinv, 0, (size_t)N * sizeof(float), stream);
    deg_kernel<<<(E + 255) / 256, 256, 0, stream>>>(dst, dinv, E);
    dinv_kernel<<<(N + 255) / 256, 256, 0, stream>>>(dinv, N);

    const int row_tiles = (N + 15) / 16;
    const int nv4  = N * 32;                 // N*128/4 float4 slots
    const int ev   = E * 32;                 // 32 lanes per edge
    const float* Wl[3] = {W0, W1, W2};
    const float* in = x;
    for (int l = 0; l < 3; ++l) {
        gemm_wmma_f32<<<row_tiles, 256, 0, stream>>>(in, Wl[l], buf0, N, l > 0);
        init_agg<<<(nv4 + 255) / 256, 256, 0, stream>>>(buf0, dinv, buf1, N);
        edge_agg<<<(ev + 255) / 256, 256, 0, stream>>>(buf0, src, dst, dinv, buf1, E);
        in = buf1;   // conv output (pre-ReLU); ReLU fused into next GEMM
    }

    // mean pool + MLP head
    hipMemsetAsync(pooled, 0, (size_t)G * DF * sizeof(float), stream);
    hipMemsetAsync(cnt,    0, (size_t)G * sizeof(float), stream);
    pool_accum<<<(nv4 + 255) / 256, 256, 0, stream>>>(buf1, bat, pooled, cnt, N);
    mlp1_kernel<<<(G * DF + 255) / 256, 256, 0, stream>>>(pooled, cnt, M0w, M0b, mlp1, G);
    mlp2_kernel<<<(G * OUT + 255) / 256, 256, 0, stream>>>(mlp1, M1w, M1b,
                                                           (float*)d_out, G, OUT);
}